// CausalConv_85521388798425
// MI455X (gfx1250) — compile-verified
//
#include <hip/hip_runtime.h>
#include <stdint.h>

// Causal depthwise conv1d, k=4, x:(8,4096,2048) f32 -> memory-bound (AI ~1 FLOP/B).
// Strategy: coalesced b128 channel-vector streaming, async global->LDS double
// buffering (ASYNCcnt pipeline), rolling 3-register time window, nontemporal
// b128 stores. No WMMA: depthwise conv has no shared-reduction matmul structure.

typedef float v4f  __attribute__((ext_vector_type(4)));
typedef int   i32x4 __attribute__((vector_size(16)));   // matches builtin param type

#define AS1 __attribute__((address_space(1)))
#define AS3 __attribute__((address_space(3)))

__device__ __forceinline__ void async_copy_b128(const float* gsrc, void* ldst) {
#if __has_builtin(__builtin_amdgcn_global_load_async_to_lds_b128)
    void* g = (void*)gsrc;                  // strip const
    AS1 i32x4* gp = (AS1 i32x4*)g;          // addrspace-cast to global int4*
    AS3 i32x4* lp = (AS3 i32x4*)ldst;       // addrspace-cast to LDS int4*
    __builtin_amdgcn_global_load_async_to_lds_b128(gp, lp, /*offset=*/0, /*cpol=*/0);
#else
    unsigned lds_off = (unsigned)(uintptr_t)ldst;  // low 32 bits of generic shared ptr = LDS byte addr
    asm volatile("global_load_async_to_lds_b128 %0, %1, off"
                 :: "v"(lds_off), "v"(gsrc) : "memory");
#endif
}

template <int N>
__device__ __forceinline__ void wait_asynccnt() {
#if __has_builtin(__builtin_amdgcn_s_wait_asynccnt)
    __builtin_amdgcn_s_wait_asynccnt(N);
#else
    asm volatile("s_wait_asynccnt %0" :: "n"(N) : "memory");
#endif
}

constexpr int B = 8, L = 4096, D = 2048;
constexpr int TPB = 256;               // 8 waves (wave32)
constexpr int VEC = 4;                 // float4 per lane
constexpr int CH_TILE = TPB * VEC;     // 1024 channels per block
constexpr int T_CHUNK = 8;             // timesteps per async chunk
constexpr int T_BLOCK = 64;            // timesteps per block tile
constexpr int NCHUNK = T_BLOCK / T_CHUNK;  // 8 chunks, double-buffered

__global__ __launch_bounds__(TPB)
void causal_conv_k4_kernel(const float* __restrict__ x,
                           const float* __restrict__ W,
                           const float* __restrict__ bias,
                           float* __restrict__ out) {
    // double-buffered staging: 2 * 8 * 256 * 16B = 64 KB LDS
    __shared__ v4f sbuf[2][T_CHUNK][TPB];

    const int tid = threadIdx.x;
    const int t0  = blockIdx.x * T_BLOCK;
    const int ch  = blockIdx.y * CH_TILE + tid * VEC;
    const int b   = blockIdx.z;

    const float* xb = x   + ((size_t)b * L) * D + ch;
    float*       ob = out + ((size_t)b * L) * D + ch;

    // Per-thread weights: 4 channels x 4 taps, contiguous 64B -> 4x b128 loads,
    // transposed in registers to per-tap channel vectors.
    const v4f* wv = (const v4f*)(W + (size_t)ch * 4);
    v4f q0 = wv[0], q1 = wv[1], q2 = wv[2], q3 = wv[3];
    v4f w0 = {q0.x, q1.x, q2.x, q3.x};   // tap for x[t-3]
    v4f w1 = {q0.y, q1.y, q2.y, q3.y};   // tap for x[t-2]
    v4f w2 = {q0.z, q1.z, q2.z, q3.z};   // tap for x[t-1]
    v4f w3 = {q0.w, q1.w, q2.w, q3.w};   // tap for x[t]
    v4f bv = *(const v4f*)(bias + ch);

    const v4f zr = {0.f, 0.f, 0.f, 0.f};

    // Causal history (t0-3..t0-1); zero only for the first time-tile.
    v4f h3 = (t0 - 3 >= 0) ? *(const v4f*)(xb + (size_t)(t0 - 3) * D) : zr;
    v4f h2 = (t0 - 2 >= 0) ? *(const v4f*)(xb + (size_t)(t0 - 2) * D) : zr;
    v4f h1 = (t0 - 1 >= 0) ? *(const v4f*)(xb + (size_t)(t0 - 1) * D) : zr;

    // Prologue: stage chunk 0 asynchronously.
    {
        const float* src = xb + (size_t)t0 * D;
#pragma unroll
        for (int ts = 0; ts < T_CHUNK; ++ts)
            async_copy_b128(src + (size_t)ts * D, &sbuf[0][ts][tid]);
    }

#pragma unroll
    for (int ci = 0; ci < NCHUNK; ++ci) {
        if (ci + 1 < NCHUNK) {
            // Issue next chunk into the other buffer, then wait until only
            // those T_CHUNK loads remain outstanding (current chunk landed).
            const float* src = xb + (size_t)(t0 + (ci + 1) * T_CHUNK) * D;
#pragma unroll
            for (int ts = 0; ts < T_CHUNK; ++ts)
                async_copy_b128(src + (size_t)ts * D, &sbuf[(ci + 1) & 1][ts][tid]);
            wait_asynccnt<T_CHUNK>();
        } else {
            wait_asynccnt<0>();
        }

#pragma unroll
        for (int ts = 0; ts < T_CHUNK; ++ts) {
            v4f xc = sbuf[ci & 1][ts][tid];                 // ds_load_b128
            v4f y  = bv + w0 * h3 + w1 * h2 + w2 * h1 + w3 * xc;
            __builtin_nontemporal_store(
                y, (v4f*)(ob + (size_t)(t0 + ci * T_CHUNK + ts) * D));
            h3 = h2; h2 = h1; h1 = xc;                      // slide causal window
        }
    }
}

extern "C" void kernel_launch(void* const* d_in, const int* in_sizes, int n_in,
                              void* d_out, int out_size, void* d_ws, size_t ws_size,
                              hipStream_t stream) {
    const float* x    = (const float*)d_in[0];
    const float* W    = (const float*)d_in[1];
    const float* bias = (const float*)d_in[2];
    float* out        = (float*)d_out;

    dim3 grid(L / T_BLOCK, D / CH_TILE, B);   // 64 x 2 x 8 = 1024 blocks
    causal_conv_k4_kernel<<<grid, TPB, 0, stream>>>(x, W, bias, out);
}